// TF2N_35347580846484
// MI455X (gfx1250) — compile-verified
//
#include <hip/hip_runtime.h>

// MI455X / gfx1250, wave32. Single-wave kernel.
//
// The reference collapses: tr = (op - op)*w == 0 for finite inputs, ti is dead,
// so real_sum == 0 and the output is Wf3·relu(Wf2·relu(bf1)+bf2)+bf3 broadcast
// to all B=32 rows. Total live work: ~66 KB of reads, ~16K MACs -> one wave.
//
// The 128x128 f32 GEMV runs on the matrix pipe via V_WMMA_F32_16X16X4_F32.

typedef __attribute__((ext_vector_type(2))) float v2f;
typedef __attribute__((ext_vector_type(8))) float v8f;

__global__ __launch_bounds__(32)
void TF2N_dead_graph_final_mlp(const float* __restrict__ bf1,
                               const float* __restrict__ Wf2,
                               const float* __restrict__ bf2,
                               const float* __restrict__ Wf3,
                               const float* __restrict__ bf3,
                               float* __restrict__ out)
{
    __shared__ float z1[128];
    __shared__ float z2[128];

    const int lane = threadIdx.x;          // 0..31, one wave32

    // ---- layer 1: z1 = relu(bf1) --------------------------------------
    for (int i = lane; i < 128; i += 32)
        z1[i] = fmaxf(bf1[i], 0.0f);
    __syncthreads();

    // ---- layer 2: z2 = relu(Wf2 @ z1 + bf2) via WMMA f32 16x16x4 ------
    // A tile (16x4 f32, ISA 7.12.2): lane<16 -> row M=lane, K = {0,1};
    //                                lane>=16 -> row M=lane-16, K = {2,3}.
    // B tile (4x16): mirrored; we only populate column N=0 with z1[k..k+3],
    // so D column 0 accumulates the GEMV result, other columns stay 0.
    // D (16x16 f32): column N=0 lives in lane 0 (M=0..7 over 8 VGPRs) and
    //                lane 16 (M=8..15 over 8 VGPRs).
    const int m    = lane & 15;
    const int koff = (lane >> 4) << 1;     // 0 for lanes 0-15, 2 for 16-31

    for (int r = 0; r < 8; ++r) {          // 8 blocks of 16 output rows
        v8f acc = {};
        const float* arow = Wf2 + (r * 16 + m) * 128 + koff;

        for (int k4 = 0; k4 < 128; k4 += 4) {
            v2f a;
            a.x = arow[k4 + 0];
            a.y = arow[k4 + 1];

            v2f b;
            b.x = 0.0f; b.y = 0.0f;
            if (lane == 0)  { b.x = z1[k4 + 0]; b.y = z1[k4 + 1]; }
            if (lane == 16) { b.x = z1[k4 + 2]; b.y = z1[k4 + 3]; }

            // (neg_a, A, neg_b, B, c_mod, C, reuse_a, reuse_b)
            acc = __builtin_amdgcn_wmma_f32_16x16x4_f32(
                      false, a, false, b, (short)0, acc, false, false);
        }

        if (lane == 0) {
            #pragma unroll
            for (int v = 0; v < 8; ++v) z2[r * 16 + v] = acc[v];
        }
        if (lane == 16) {
            #pragma unroll
            for (int v = 0; v < 8; ++v) z2[r * 16 + 8 + v] = acc[v];
        }
    }
    __syncthreads();

    for (int i = lane; i < 128; i += 32)
        z2[i] = fmaxf(z2[i] + bf2[i], 0.0f);
    __syncthreads();

    // ---- layer 3: scalar = Wf3 . z2 + bf3, broadcast to 32 outputs ----
    float s = 0.0f;
    for (int i = lane; i < 128; i += 32)
        s += Wf3[i] * z2[i];

    #pragma unroll
    for (int off = 16; off > 0; off >>= 1)
        s += __shfl_down(s, off, 32);
    s = __shfl(s, 0, 32);                  // broadcast lane 0's total

    out[lane] = s + bf3[0];                // B=32 identical outputs
}

extern "C" void kernel_launch(void* const* d_in, const int* in_sizes, int n_in,
                              void* d_out, int out_size, void* d_ws, size_t ws_size,
                              hipStream_t stream) {
    (void)in_sizes; (void)n_in; (void)out_size; (void)d_ws; (void)ws_size;
    // setup_inputs() order: ... 26:Wf1 27:bf1 28:Wf2 29:bf2 30:Wf3 31:bf3
    const float* bf1 = (const float*)d_in[27];
    const float* Wf2 = (const float*)d_in[28];
    const float* bf2 = (const float*)d_in[29];
    const float* Wf3 = (const float*)d_in[30];
    const float* bf3 = (const float*)d_in[31];

    hipLaunchKernelGGL(TF2N_dead_graph_final_mlp,
                       dim3(1), dim3(32), 0, stream,
                       bf1, Wf2, bf2, Wf3, bf3, (float*)d_out);
}